// Butterfly_34359738515
// MI455X (gfx1250) — compile-verified
//
#include <hip/hip_runtime.h>
#include <cstddef>
#include <cstdint>

typedef __attribute__((ext_vector_type(2))) float v2f;
typedef __attribute__((ext_vector_type(8))) float v8f;

#define PITCH 1028   // floats per LDS row: 1024 + 4 pad; 16B-aligned rows, de-phases banks

// ---------------------------------------------------------------------------
// Precompute kernel: collapse butterfly stages into dense small matrices.
//   blocks 0..63    : MA[seg]   = stage3*stage2*stage1*stage0 restricted to segment seg (16x16)
//   blocks 64..127  : MB[combo] = stage7*..*stage4 on k-positions at (block256=combo>>4, lane=combo&15)
//   block 128       : MC[p]     = stage9*stage8 on the 4 quarters at position p (4x4)
// twiddle layout: tw[idx*2048 + pair*4 + r*2 + c], pair = (g0/(2*stride))*stride + g0%stride
// ---------------------------------------------------------------------------
__global__ __launch_bounds__(256) void butterfly_prep(const float* __restrict__ tw,
                                                      float* __restrict__ MA,
                                                      float* __restrict__ MB,
                                                      float* __restrict__ MC) {
  __shared__ float Mbuf0[256];
  __shared__ float Mbuf1[256];
  const int tid = threadIdx.x;
  const int bid = blockIdx.x;

  if (bid < 128) {
    const int r = tid >> 4;      // output row of 16x16
    const int c = tid & 15;      // input col
    const bool isA = (bid < 64);
    float* cur = Mbuf0;
    float* nxt = Mbuf1;
    cur[tid] = (r == c) ? 1.0f : 0.0f;
    __syncthreads();

    for (int s = 0; s < 4; ++s) {
      const int idx = isA ? s : (s + 4);
      const int stride = 1 << idx;                 // global stride
      const int sl = isA ? stride : (stride >> 4); // local stride on the 16-row index
      const int pos = r & (2 * sl - 1);
      const int rsel = (pos < sl) ? 0 : 1;
      const int k0 = r - rsel * sl;                // local i0
      const int k1 = k0 + sl;                      // local i1
      int g0;                                      // global element index of i0
      if (isA) {
        g0 = bid * 16 + k0;
      } else {
        const int combo = bid - 64;
        g0 = (combo >> 4) * 256 + k0 * 16 + (combo & 15);
      }
      const int pair = (g0 / (2 * stride)) * stride + (g0 % stride);
      const float* t = tw + (size_t)idx * 2048 + (size_t)pair * 4 + rsel * 2;
      const float nv = t[0] * cur[k0 * 16 + c] + t[1] * cur[k1 * 16 + c];
      nxt[r * 16 + c] = nv;
      __syncthreads();
      float* tmp = cur; cur = nxt; nxt = tmp;
    }
    if (isA) MA[bid * 256 + tid] = cur[tid];
    else     MB[(bid - 64) * 256 + tid] = cur[tid];
  } else {
    // MC: one thread per position p, build 4x4 = stage9 * stage8
    const int p = tid;
    float M8[4][4];
#pragma unroll
    for (int q = 0; q < 4; ++q)
#pragma unroll
      for (int c2 = 0; c2 < 4; ++c2) M8[q][c2] = 0.0f;
#pragma unroll
    for (int q = 0; q < 4; ++q) {               // stage 8: stride 256 -> local stride 1 on q
      const int rsel = q & 1;
      const int q0 = q - rsel;
      const int pair = (q0 >> 1) * 256 + p;
      const float* t = tw + (size_t)8 * 2048 + (size_t)pair * 4 + rsel * 2;
      M8[q][q0] = t[0];
      M8[q][q0 + 1] = t[1];
    }
    float Mf[4][4];
#pragma unroll
    for (int q = 0; q < 4; ++q) {               // stage 9: stride 512 -> local stride 2 on q
      const int rsel = q >> 1;
      const int q0 = q - 2 * rsel;
      const int pair = q0 * 256 + p;
      const float* t = tw + (size_t)9 * 2048 + (size_t)pair * 4 + rsel * 2;
#pragma unroll
      for (int c2 = 0; c2 < 4; ++c2)
        Mf[q][c2] = t[0] * M8[q0][c2] + t[1] * M8[q0 + 2][c2];
    }
#pragma unroll
    for (int q = 0; q < 4; ++q)
#pragma unroll
      for (int c2 = 0; c2 < 4; ++c2)
        MC[p * 16 + q * 4 + c2] = Mf[q][c2];
  }
}

// ---------------------------------------------------------------------------
// Main fused kernel: 16 batch rows x 1024 elements per workgroup, all 10
// stages resident in LDS.
//   stage-in : GLOBAL_LOAD_ASYNC_TO_LDS_B128 (ASYNCcnt DMA path, no VGPR trip)
//   Phase A/B: 16x16 matmuls on matrix units (4 chained v_wmma_f32_16x16x4_f32)
//   Phase C  : 4x4 per position on VALU, bias folded in, direct coalesced store
// LDS layout: xs[row * PITCH + elem] (row-major; async b128 needs contiguity).
// ---------------------------------------------------------------------------
__global__ __launch_bounds__(256) void butterfly_main(const float* __restrict__ x,
                                                      const float* __restrict__ bias,
                                                      const float* __restrict__ MA,
                                                      const float* __restrict__ MB,
                                                      const float* __restrict__ MC,
                                                      float* __restrict__ out) {
  __shared__ float xs[16 * PITCH];
  const int tid = threadIdx.x;
  const int lane = tid & 31;
  const int wave = tid >> 5;       // 0..7
  const int hi = lane >> 4;        // 0 or 1 (lane half)
  const int ln16 = lane & 15;
  const long r0 = (long)blockIdx.x * 16;

  // ---- Stage in: 16 rows x 1024 f32 via async global->LDS DMA (16B/lane) ----
  {
    const uint64_t gbase = (uint64_t)(uintptr_t)(x + r0 * 1024);
#pragma unroll 4
    for (int i = 0; i < 16; ++i) {
      const int lin = i * 256 + tid;              // 0..4095 b128 chunks
      const int row = lin >> 8;                   // 0..15
      const int e4 = (lin & 255) << 2;            // 0..1020
      const unsigned goff = (unsigned)((row * 1024 + e4) * 4);      // byte offset
      const unsigned laddr = (unsigned)(uintptr_t)&xs[row * PITCH + e4]; // LDS byte addr
      asm volatile("global_load_async_to_lds_b128 %0, %1, %2"
                   :: "v"(laddr), "v"(goff), "s"(gbase) : "memory");
    }
    asm volatile("s_wait_asynccnt 0x0" ::: "memory");
  }
  __syncthreads();

  // ---- Phase A: stages 0-3, one 16x16 matmul per contiguous 16-segment ----
  for (int j = 0; j < 8; ++j) {
    const int seg = wave + 8 * j;         // each wave owns disjoint segments
    const float* mp = MA + seg * 256 + ln16 * 16 + (hi << 1);
    v8f acc = {};
#pragma unroll
    for (int k = 0; k < 4; ++k) {
      v2f a; a.x = mp[4 * k]; a.y = mp[4 * k + 1];
      const int e = seg * 16 + 4 * k + (hi << 1);
      v2f b; b.x = xs[ln16 * PITCH + e]; b.y = xs[ln16 * PITCH + e + 1];
      acc = __builtin_amdgcn_wmma_f32_16x16x4_f32(false, a, false, b,
                                                  (short)0, acc, false, false);
    }
#pragma unroll
    for (int rr = 0; rr < 8; ++rr) {
      xs[ln16 * PITCH + seg * 16 + rr + 8 * hi] = acc[rr]; // in-place: seg exclusive to wave
    }
  }
  __syncthreads();

  // ---- Phase B: stages 4-7, 16x16 matmul across k at fixed (block256, lane) ----
  for (int j = 0; j < 8; ++j) {
    const int combo = wave + 8 * j;
    const int bblk = combo >> 4;          // 0..3
    const int lpos = combo & 15;          // 0..15
    const float* mp = MB + combo * 256 + ln16 * 16 + (hi << 1);
    v8f acc = {};
#pragma unroll
    for (int k = 0; k < 4; ++k) {
      v2f a; a.x = mp[4 * k]; a.y = mp[4 * k + 1];
      const int e = bblk * 256 + (4 * k + (hi << 1)) * 16 + lpos;
      v2f b; b.x = xs[ln16 * PITCH + e]; b.y = xs[ln16 * PITCH + e + 16];
      acc = __builtin_amdgcn_wmma_f32_16x16x4_f32(false, a, false, b,
                                                  (short)0, acc, false, false);
    }
#pragma unroll
    for (int rr = 0; rr < 8; ++rr) {
      xs[ln16 * PITCH + bblk * 256 + (rr + 8 * hi) * 16 + lpos] = acc[rr];
    }
  }
  __syncthreads();

  // ---- Phase C: stages 8-9 (4x4 per position) + bias, store straight out ----
  {
    const int p = tid;                    // 0..255
    const float* m = MC + p * 16;
    const float m0 = m[0],  m1 = m[1],  m2 = m[2],  m3 = m[3];
    const float m4 = m[4],  m5 = m[5],  m6 = m[6],  m7 = m[7];
    const float m8 = m[8],  m9 = m[9],  mA = m[10], mB2 = m[11];
    const float mC2 = m[12], mD = m[13], mE = m[14], mF = m[15];
    const float b0 = bias[p], b1 = bias[p + 256], b2 = bias[p + 512], b3 = bias[p + 768];
#pragma unroll 4
    for (int row = 0; row < 16; ++row) {
      const float v0 = xs[row * PITCH + p];
      const float v1 = xs[row * PITCH + p + 256];
      const float v2 = xs[row * PITCH + p + 512];
      const float v3 = xs[row * PITCH + p + 768];
      float* op = out + (r0 + row) * 1024 + p;    // lane-consecutive -> coalesced
      op[0]   = m0 * v0 + m1 * v1 + m2 * v2 + m3 * v3 + b0;
      op[256] = m4 * v0 + m5 * v1 + m6 * v2 + m7 * v3 + b1;
      op[512] = m8 * v0 + m9 * v1 + mA * v2 + mB2 * v3 + b2;
      op[768] = mC2 * v0 + mD * v1 + mE * v2 + mF * v3 + b3;
    }
  }
}

extern "C" void kernel_launch(void* const* d_in, const int* in_sizes, int n_in,
                              void* d_out, int out_size, void* d_ws, size_t ws_size,
                              hipStream_t stream) {
  (void)n_in; (void)out_size; (void)ws_size;
  const float* x    = (const float*)d_in[0];   // (batch, 1024) f32
  const float* tw   = (const float*)d_in[1];   // (1, 10, 512, 2, 2) f32
  const float* bias = (const float*)d_in[2];   // (1024,) f32
  float* out = (float*)d_out;

  float* MA = (float*)d_ws;          // 64 * 256 f32
  float* MB = MA + 64 * 256;         // 64 * 256 f32
  float* MC = MB + 64 * 256;         // 256 * 16 f32   (total 144 KB)

  const int batch = in_sizes[0] / 1024;

  butterfly_prep<<<129, 256, 0, stream>>>(tw, MA, MB, MC);
  butterfly_main<<<batch / 16, 256, 0, stream>>>(x, bias, MA, MB, MC, out);
}